// SAGE_OGB_90726889160781
// MI455X (gfx1250) — compile-verified
//
#include <hip/hip_runtime.h>
#include <math.h>

typedef float v8f __attribute__((ext_vector_type(8)));
typedef float v2f __attribute__((ext_vector_type(2)));

#define D_FEAT 128      // D == H == 128 for this model
#define TILE_COLS 64    // 4 x 16-wide column tiles per block

// ---------------------------------------------------------------- utilities
__global__ void fill_f32(float* p, float v, long n) {
    long i = (long)blockIdx.x * blockDim.x + threadIdx.x;
    if (i < n) p[i] = v;
}

__global__ void count_edges(const int* __restrict__ dst, float* __restrict__ cnt, int E) {
    int i = blockIdx.x * blockDim.x + threadIdx.x;
    if (i < E) atomicAdd(&cnt[dst[i]], 1.0f);
}

__global__ void inv_count(const float* __restrict__ cnt, float* __restrict__ invc, int N) {
    int i = blockIdx.x * blockDim.x + threadIdx.x;
    if (i < N) invc[i] = 1.0f / fmaxf(cnt[i], 1.0f);
}

// ------------------------------------------------------- edge scatter (mean numerator)
// one wave per edge; 32 lanes stride the 128 features (4 atomics/lane).
// Node feature matrix (51MB) fits in MI455X's 192MB L2, so gathers hit L2.
__global__ void scatter_add(const float* __restrict__ x,
                            const int* __restrict__ src, const int* __restrict__ dst,
                            float* __restrict__ agg, int E) {
    int wave = (int)((blockIdx.x * (long)blockDim.x + threadIdx.x) >> 5);
    int lane = threadIdx.x & 31;
    if (wave >= E) return;
    int s = src[wave], d = dst[wave];
    const float* xs = x + (size_t)s * D_FEAT;
    float* ad = agg + (size_t)d * D_FEAT;
#pragma unroll
    for (int f = 0; f < D_FEAT; f += 32)
        atomicAdd(&ad[f + lane], xs[f + lane]);
}

// ------------------------------------------------------- fused dual GEMM via WMMA f32
// out[r,c] = (agg[r,:] * invc[r]) @ Wl + bias[c] + xin[r,:] @ Wr
// Weight slabs (128 x 64 cols, both Wl and Wr) are staged into LDS with CDNA5
// async-to-LDS copies (ASYNCcnt + s_wait_asynccnt), then each of 4 waves per
// block owns one 16x16 tile and runs K=128 as 32x2 V_WMMA_F32_16X16X4_F32.
__global__ void sage_gemm_wmma(const float* __restrict__ agg,
                               const float* __restrict__ invc,
                               const float* __restrict__ xin,
                               const float* __restrict__ Wl,
                               const float* __restrict__ bias,
                               const float* __restrict__ Wr,
                               float* __restrict__ out,
                               int N, int C, int colTiles) {
    __shared__ float sWl[D_FEAT * TILE_COLS];
    __shared__ float sWr[D_FEAT * TILE_COLS];

    const int c0  = blockIdx.y * TILE_COLS;
    const int tid = threadIdx.y * 32 + threadIdx.x;   // 0..127, all active

    // ---- stage weight slabs into LDS via async copies (EXEC all-1s) ----
    for (int idx = tid; idx < D_FEAT * TILE_COLS; idx += 128) {
        int k  = idx >> 6;          // 0..127
        int cc = idx & 63;          // 0..63
        int gc = c0 + cc;
        int gcc = gc < C ? gc : 0;  // clamp; masked to 0 at fragment build
        const float* gl = Wl + (size_t)k * C + gcc;
        const float* gr = Wr + (size_t)k * C + gcc;
        unsigned la = (unsigned)(size_t)&sWl[idx];
        unsigned lb = (unsigned)(size_t)&sWr[idx];
        asm volatile("global_load_async_to_lds_b32 %0, %1, off"
                     :: "v"(la), "v"(gl) : "memory");
        asm volatile("global_load_async_to_lds_b32 %0, %1, off"
                     :: "v"(lb), "v"(gr) : "memory");
    }
    asm volatile("s_wait_asynccnt 0x0" ::: "memory");
    __syncthreads();

    const int colTile = blockIdx.y * 4 + threadIdx.y;
    if (colTile < colTiles) {       // wave-uniform: EXEC stays all-1s for WMMA
        const int lane = threadIdx.x;   // 0..31
        const int half = lane >> 4;     // A: K+0/1 vs K+2/3 ; C/D: M base
        const int l    = lane & 15;

        // A fragment rows (mean-divide fused via invc)
        int row = blockIdx.x * 16 + l;
        int rowIdx = row < N ? row : N - 1;
        float rowMask = row < N ? 1.0f : 0.0f;
        float sc = invc[rowIdx] * rowMask;
        const float2* aRow = reinterpret_cast<const float2*>(agg) + (size_t)rowIdx * (D_FEAT / 2);
        const float2* xRow = reinterpret_cast<const float2*>(xin) + (size_t)rowIdx * (D_FEAT / 2);

        // B fragment cols (from LDS slab)
        int ccol = threadIdx.y * 16 + l;     // 0..63 within slab
        int col  = c0 + ccol;
        float colMask = col < C ? 1.0f : 0.0f;

        v8f acc = {};
#pragma unroll 4
        for (int k = 0; k < D_FEAT; k += 4) {
            int ka = k + half * 2;           // even K pair for this lane half
            float2 a2 = aRow[ka >> 1];
            float2 x2 = xRow[ka >> 1];
            v2f aA = { a2.x * sc,      a2.y * sc };
            v2f aX = { x2.x * rowMask, x2.y * rowMask };
            v2f bL = { sWl[ka * TILE_COLS + ccol] * colMask,
                       sWl[(ka + 1) * TILE_COLS + ccol] * colMask };
            v2f bR = { sWr[ka * TILE_COLS + ccol] * colMask,
                       sWr[(ka + 1) * TILE_COLS + ccol] * colMask };
            acc = __builtin_amdgcn_wmma_f32_16x16x4_f32(false, aA, false, bL,
                                                        (short)0, acc, false, false);
            acc = __builtin_amdgcn_wmma_f32_16x16x4_f32(false, aX, false, bR,
                                                        (short)0, acc, false, false);
        }

        // C/D layout: VGPR r -> M = 16*tile + r + 8*half, N = l
        float b = (col < C) ? bias[col] : 0.0f;
#pragma unroll
        for (int r = 0; r < 8; ++r) {
            int om = blockIdx.x * 16 + r + half * 8;
            if (om < N && col < C)
                out[(size_t)om * C + col] = acc[r] + b;
        }
    }
}

// ------------------------------------------------------- batchnorm stats + apply
__global__ void bn_stats(const float* __restrict__ h, float* __restrict__ stats, int N) {
    int col = threadIdx.x;                        // blockDim.x == 128
    float s = 0.0f, s2 = 0.0f;
    for (int r = blockIdx.x; r < N; r += gridDim.x) {
        float v = h[(size_t)r * D_FEAT + col];
        s += v; s2 += v * v;
    }
    atomicAdd(&stats[col], s);
    atomicAdd(&stats[D_FEAT + col], s2);
}

__global__ void bn_relu(float* __restrict__ h, const float* __restrict__ stats,
                        const float* __restrict__ g, const float* __restrict__ b,
                        int N) {
    long i = (long)blockIdx.x * blockDim.x + threadIdx.x;
    if (i >= (long)N * D_FEAT) return;
    int col = (int)(i & (D_FEAT - 1));
    float invN = 1.0f / (float)N;
    float mu = stats[col] * invN;
    float var = stats[D_FEAT + col] * invN - mu * mu;
    float v = (h[i] - mu) * rsqrtf(var + 1e-5f) * g[col] + b[col];
    h[i] = fmaxf(v, 0.0f);
}

// ------------------------------------------------------- row log-softmax (wave per row)
__global__ void log_softmax_rows(float* __restrict__ out, int N, int C) {
    int row = blockIdx.x * (blockDim.x >> 5) + (threadIdx.x >> 5);
    int lane = threadIdx.x & 31;
    if (row >= N) return;
    float* p = out + (size_t)row * C;
    float a = (lane < C)      ? p[lane]      : -INFINITY;
    float b = (lane + 32 < C) ? p[lane + 32] : -INFINITY;
    float m = fmaxf(a, b);
#pragma unroll
    for (int o = 16; o; o >>= 1) m = fmaxf(m, __shfl_xor(m, o, 32));
    float e = ((lane < C) ? __expf(a - m) : 0.0f) +
              ((lane + 32 < C) ? __expf(b - m) : 0.0f);
#pragma unroll
    for (int o = 16; o; o >>= 1) e += __shfl_xor(e, o, 32);
    float lse = m + __logf(e);
    if (lane < C)      p[lane]      = a - lse;
    if (lane + 32 < C) p[lane + 32] = b - lse;
}

// ---------------------------------------------------------------- driver
extern "C" void kernel_launch(void* const* d_in, const int* in_sizes, int n_in,
                              void* d_out, int out_size, void* d_ws, size_t ws_size,
                              hipStream_t stream) {
    const float* x   = (const float*)d_in[0];
    const int*   ei  = (const int*)d_in[1];
    const float* Wl0 = (const float*)d_in[2];
    const float* bl0 = (const float*)d_in[3];
    const float* Wr0 = (const float*)d_in[4];
    const float* g0  = (const float*)d_in[5];
    const float* be0 = (const float*)d_in[6];
    const float* Wl1 = (const float*)d_in[7];
    const float* bl1 = (const float*)d_in[8];
    const float* Wr1 = (const float*)d_in[9];
    const float* g1  = (const float*)d_in[10];
    const float* be1 = (const float*)d_in[11];
    const float* Wl2 = (const float*)d_in[12];
    const float* bl2 = (const float*)d_in[13];
    const float* Wr2 = (const float*)d_in[14];
    float* out = (float*)d_out;

    const int N = in_sizes[0] / D_FEAT;
    const int E = in_sizes[1] / 2;
    const int C = in_sizes[13];              // 47
    const int* src = ei;
    const int* dst = ei + E;

    // workspace layout
    char* ws = (char*)d_ws;
    float* agg   = (float*)ws;                              ws += (size_t)N * D_FEAT * 4;
    float* h0    = (float*)ws;                              ws += (size_t)N * D_FEAT * 4;
    float* h1    = (float*)ws;                              ws += (size_t)N * D_FEAT * 4;
    float* cnt   = (float*)ws;                              ws += (size_t)N * 4;
    float* invc  = (float*)ws;                              ws += (size_t)N * 4;
    float* stats = (float*)ws;                              ws += 2 * D_FEAT * 4;

    const long ND = (long)N * D_FEAT;
    const int TB = 256;
    dim3 gemmBlk(32, 4);
    int rowTiles = (N + 15) / 16;

    // degree -> inverse count (edges fixed across layers)
    fill_f32<<<(int)((N + TB - 1) / TB), TB, 0, stream>>>(cnt, 0.0f, N);
    count_edges<<<(E + TB - 1) / TB, TB, 0, stream>>>(dst, cnt, E);
    inv_count<<<(N + TB - 1) / TB, TB, 0, stream>>>(cnt, invc, N);

    // ---- layer 0
    fill_f32<<<(int)((ND + TB - 1) / TB), TB, 0, stream>>>(agg, 0.0f, ND);
    scatter_add<<<(int)(((long)E * 32 + TB - 1) / TB), TB, 0, stream>>>(x, src, dst, agg, E);
    {
        int colTiles = D_FEAT / 16;
        dim3 grid(rowTiles, (colTiles + 3) / 4);
        sage_gemm_wmma<<<grid, gemmBlk, 0, stream>>>(agg, invc, x, Wl0, bl0, Wr0,
                                                     h0, N, D_FEAT, colTiles);
    }
    fill_f32<<<1, 256, 0, stream>>>(stats, 0.0f, 2 * D_FEAT);
    bn_stats<<<960, D_FEAT, 0, stream>>>(h0, stats, N);
    bn_relu<<<(int)((ND + TB - 1) / TB), TB, 0, stream>>>(h0, stats, g0, be0, N);

    // ---- layer 1
    fill_f32<<<(int)((ND + TB - 1) / TB), TB, 0, stream>>>(agg, 0.0f, ND);
    scatter_add<<<(int)(((long)E * 32 + TB - 1) / TB), TB, 0, stream>>>(h0, src, dst, agg, E);
    {
        int colTiles = D_FEAT / 16;
        dim3 grid(rowTiles, (colTiles + 3) / 4);
        sage_gemm_wmma<<<grid, gemmBlk, 0, stream>>>(agg, invc, h0, Wl1, bl1, Wr1,
                                                     h1, N, D_FEAT, colTiles);
    }
    fill_f32<<<1, 256, 0, stream>>>(stats, 0.0f, 2 * D_FEAT);
    bn_stats<<<960, D_FEAT, 0, stream>>>(h1, stats, N);
    bn_relu<<<(int)((ND + TB - 1) / TB), TB, 0, stream>>>(h1, stats, g1, be1, N);

    // ---- layer 2 (C = 47, guarded stores straight into d_out)
    fill_f32<<<(int)((ND + TB - 1) / TB), TB, 0, stream>>>(agg, 0.0f, ND);
    scatter_add<<<(int)(((long)E * 32 + TB - 1) / TB), TB, 0, stream>>>(h1, src, dst, agg, E);
    {
        int colTiles = (C + 15) / 16;
        dim3 grid(rowTiles, (colTiles + 3) / 4);
        sage_gemm_wmma<<<grid, gemmBlk, 0, stream>>>(agg, invc, h1, Wl2, bl2, Wr2,
                                                     out, N, C, colTiles);
    }
    log_softmax_rows<<<(N + 7) / 8, 256, 0, stream>>>(out, N, C);
}